// Unsigned_MSELoss_52364241272971
// MI455X (gfx1250) — compile-verified
//
#include <hip/hip_runtime.h>

// ---------------------------------------------------------------------------
// Problem constants (from the reference): N=131072 rows, D=256 cols, B=16 segs
// Memory-bound: 256 MB f32 reads / 23.3 TB/s ~= 11.5 us floor.
// Strategy: one wave32 owns a 16-row tile. Per-lane f32 FMA accumulation of
// squared diffs/sums over the 256 columns, then ONE V_WMMA_F32_16X16X4_F32
// (full f32 precision) with an all-ones B matrix collapses the 16x4 per-lane
// partial grid into the 16 row sums. Sliced global f32 atomics -> tiny
// finalize kernel computes sum_b min(loss_pos, loss_neg) with tie -> pos.
// ---------------------------------------------------------------------------

#define NROWS  131072
#define DCOLS  256
#define BSEG   16
#define SLICES 128
#define NTILES (NROWS / 16)

typedef __attribute__((ext_vector_type(2))) float v2f;
typedef __attribute__((ext_vector_type(8))) float v8f;

__global__ __launch_bounds__(256) void
seg_sq_wmma_kernel(const float* __restrict__ inp,
                   const float* __restrict__ tgt,
                   const int*   __restrict__ batch,
                   float*       __restrict__ ws /* [2][BSEG][SLICES] */) {
    const int lane         = threadIdx.x & 31;
    const int waveInBlock  = threadIdx.x >> 5;
    const int wavesPerBlk  = blockDim.x >> 5;
    const int gwave        = blockIdx.x * wavesPerBlk + waveInBlock;
    const int nwaves       = gridDim.x * wavesPerBlk;

    // A-matrix 16x4 f32 lane mapping (ISA 7.12.2):
    //   lanes 0-15 : M = lane,     VGPR0 -> K=0, VGPR1 -> K=1
    //   lanes 16-31: M = lane-16,  VGPR0 -> K=2, VGPR1 -> K=3
    const int m     = lane & 15;
    const int kbase = (lane >> 4) << 1;          // 0 or 2

    for (int tile = gwave; tile < NTILES; tile += nwaves) {
        const int row0 = tile << 4;
        const float* pin = inp + (size_t)(row0 + m) * DCOLS + kbase;
        const float* ptg = tgt + (size_t)(row0 + m) * DCOLS + kbase;

        float ap0 = 0.f, ap1 = 0.f;              // A_pos: K = kbase, kbase+1
        float an0 = 0.f, an1 = 0.f;              // A_neg

        // 64 column chunks of 4; each lane reads 8 B of input + 8 B of target.
        // Lines are consumed sequentially -> every 128B line fetched once,
        // L0 covers the 8 revisits; 4 KB live working set per wave.
        #pragma unroll 8
        for (int c = 0; c < DCOLS; c += 4) {
            float2 a = *(const float2*)(pin + c);
            float2 b = *(const float2*)(ptg + c);
            float d0 = a.x - b.x, d1 = a.y - b.y;
            float s0 = a.x + b.x, s1 = a.y + b.y;
            ap0 = fmaf(d0, d0, ap0);
            ap1 = fmaf(d1, d1, ap1);
            an0 = fmaf(s0, s0, an0);
            an1 = fmaf(s1, s1, an1);
        }

        v2f Apos = {ap0, ap1};
        v2f Aneg = {an0, an1};
        v2f Bones = {1.0f, 1.0f};                // all-ones B: D[m][n] = sum_k A[m][k]
        v8f c0 = {};

        // Full-precision f32 WMMA: exact cross-lane 16x4 -> 16 row-sum reduce.
        v8f dpos = __builtin_amdgcn_wmma_f32_16x16x4_f32(
            false, Apos, false, Bones, (short)0, c0, false, false);
        v8f dneg = __builtin_amdgcn_wmma_f32_16x16x4_f32(
            false, Aneg, false, Bones, (short)0, c0, false, false);

        // C/D layout: VGPR j, lanes 0-15 -> M=j, lanes 16-31 -> M=j+8.
        // Lanes with N==0 (lane 0 and lane 16) publish the 16 row sums.
        if ((lane & 15) == 0) {
            const int mBase = (lane >> 4) << 3;  // 0 or 8
            const int slice = gwave & (SLICES - 1);
            #pragma unroll
            for (int j = 0; j < 8; ++j) {
                const int r   = row0 + mBase + j;
                const int seg = batch[r];
                atomicAdd(&ws[(0 * BSEG + seg) * SLICES + slice], dpos[j]);
                atomicAdd(&ws[(1 * BSEG + seg) * SLICES + slice], dneg[j]);
            }
        }
    }
}

__global__ __launch_bounds__(64) void
finalize_kernel(const float* __restrict__ ws, float* __restrict__ out) {
    __shared__ float red[BSEG];
    const int t = threadIdx.x;
    if (t < BSEG) {
        float p = 0.f, n = 0.f;
        for (int s = 0; s < SLICES; ++s) {
            p += ws[(0 * BSEG + t) * SLICES + s];
            n += ws[(1 * BSEG + t) * SLICES + s];
        }
        red[t] = (p <= n) ? p : n;               // tie -> +1 branch (loss_pos)
    }
    __syncthreads();
    if (t == 0) {
        float acc = 0.f;
        #pragma unroll
        for (int j = 0; j < BSEG; ++j) acc += red[j];
        out[0] = acc;
    }
}

extern "C" void kernel_launch(void* const* d_in, const int* in_sizes, int n_in,
                              void* d_out, int out_size, void* d_ws, size_t ws_size,
                              hipStream_t stream) {
    const float* inp   = (const float*)d_in[0];
    const float* tgt   = (const float*)d_in[1];
    const int*   batch = (const int*)d_in[2];   // int32 (JAX default x64-off)
    float*       ws    = (float*)d_ws;

    const size_t wsBytes = (size_t)2 * BSEG * SLICES * sizeof(float);
    hipMemsetAsync(d_ws, 0, wsBytes, stream);   // graph-capture-safe zeroing

    // 512 blocks x 8 waves = 4096 waves; 2 tiles per wave over 8192 tiles.
    seg_sq_wmma_kernel<<<512, 256, 0, stream>>>(inp, tgt, batch, ws);
    finalize_kernel<<<1, 64, 0, stream>>>(ws, (float*)d_out);
}